// PiecewiseShared_70952859730191
// MI455X (gfx1250) — compile-verified
//
#include <hip/hip_runtime.h>
#include <stdint.h>

#define SEGMENTS 32
#define C_CH     64
#define NNODES   4096
#define BATCH    32
#define WROW     97     // (N_BASIS-1)*SEGMENTS+1
#define NP       2      // row-tiles (p values) per block
#define ROWPAD   (65 * 64)   // dwords per padded 4096-dword row (1 pad dword / 64)

typedef __attribute__((ext_vector_type(4))) unsigned int v4u;
typedef __attribute__((ext_vector_type(8))) int          v8i;
typedef __attribute__((ext_vector_type(4))) int          v4i;

// Flat (generic) pointers into LDS carry the LDS byte address in their low 32
// bits (aperture in the high bits) -> truncate to get D#.lds_addr.
static __device__ __forceinline__ unsigned lds_addr_of(const void* p) {
  return (unsigned)(unsigned long long)p;
}

// D# group 0: [1:0]=count=1, [63:32]=lds_addr, [120:64]=global_addr, [127:126]=type=2
static __device__ __forceinline__ v4u tdm_group0(unsigned lds_byte_addr,
                                                 unsigned long long gaddr) {
  v4u g;
  g.x = 0x1u;
  g.y = lds_byte_addr;
  g.z = (unsigned)(gaddr & 0xffffffffull);
  g.w = (unsigned)((gaddr >> 32) & 0x1ffffffull) | (2u << 30);
  return g;
}

// D# group 1 for a <=2D tile of 4-byte elements (data_size=2), no multicast,
// optional LDS padding (pad every (2<<interval) dwords by (amount+1) dwords).
static __device__ __forceinline__ v8i tdm_group1(unsigned tensor_dim0,
                                                 unsigned tensor_dim1,
                                                 unsigned tile_dim0,
                                                 unsigned tile_dim1,
                                                 unsigned long long stride0,
                                                 unsigned pad_enable,
                                                 unsigned pad_interval,
                                                 unsigned pad_amount) {
  unsigned w0 = (2u << 16) | ((pad_enable & 1u) << 20) |
                ((pad_interval & 7u) << 22) | ((pad_amount & 127u) << 25);
  v8i g;
  g[0] = (int)w0;
  g[1] = (int)((tensor_dim0 & 0xffffu) << 16);                                   // barrier addr = 0
  g[2] = (int)(((tensor_dim0 >> 16) & 0xffffu) | ((tensor_dim1 & 0xffffu) << 16));
  g[3] = (int)(((tensor_dim1 >> 16) & 0xffffu) | ((tile_dim0 & 0xffffu) << 16));
  g[4] = (int)(tile_dim1 & 0xffffu);                                             // tile_dim2 = 0
  g[5] = (int)(unsigned)(stride0 & 0xffffffffull);
  g[6] = (int)(unsigned)((stride0 >> 32) & 0xffffull);                           // dim1_stride = 0
  g[7] = 0;
  return g;
}

static __device__ __forceinline__ void tdm_load(v4u g0, v8i g1) {
  v4i z = {0, 0, 0, 0};
#if __clang_major__ >= 23
  v8i z8 = {0, 0, 0, 0, 0, 0, 0, 0};
  __builtin_amdgcn_tensor_load_to_lds(g0, g1, z, z, z8, 0);
#else
  __builtin_amdgcn_tensor_load_to_lds(g0, g1, z, z, 0);
#endif
}

__global__ __launch_bounds__(256)
void PiecewiseShared_70952859730191_kernel(const float* __restrict__ x,
                                           const float* __restrict__ w,
                                           float* __restrict__ out) {
  // NP rows x[b, p_base+j, :], each 4096 dwords + 1 pad dword per 64 dwords.
  // Row length = exactly 64 pad intervals, so row j starts at dword j*4160 and
  // element k of a row sits at q*65 + c (k = q*64 + c) -> transposed reads are
  // bank-conflict-free (bank = (q + c) mod 64, q varies across lanes).
  __shared__ float s_row[NP * ROWPAD];    // 33280 B
  __shared__ float s_w[C_CH * WROW];      // 24832 B  (58112 B total, < 64 KB)

  const int pb = blockIdx.x * NP;         // first node-block of this workgroup
  const int b  = blockIdx.y;              // batch 0..31
  const int t  = threadIdx.x;

  if (t < 32) {                           // wave 0 drives the TDM (EXEC ignored by TDM)
    // One 2D TDM for all NP rows, padded into LDS.
    unsigned long long rbase =
        (unsigned long long)(x + ((size_t)b * C_CH + (size_t)pb) * NNODES);
    v4u g0 = tdm_group0(lds_addr_of(s_row), rbase);
    v8i g1 = tdm_group1(/*tdim0=*/NNODES, /*tdim1=*/NP,
                        /*tile0=*/NNODES, /*tile1=*/NP,
                        /*stride0=*/NNODES,
                        /*pad_en=*/1, /*pad_interval=*/5 /*64 dwords*/,
                        /*pad_amount=*/0 /*1 dword*/);
    tdm_load(g0, g1);

    // Whole weight table w[64,97] (amortized over NP tiles).
    v4u h0 = tdm_group0(lds_addr_of(s_w), (unsigned long long)w);
    v8i h1 = tdm_group1(/*tdim0=*/WROW, /*tdim1=*/C_CH,
                        /*tile0=*/WROW, /*tile1=*/C_CH,
                        /*stride0=*/WROW, 0, 0, 0);
    tdm_load(h0, h1);

    __builtin_amdgcn_s_wait_tensorcnt(0);
  }
  __syncthreads();

#pragma unroll
  for (int j = 0; j < NP; ++j) {
    const int p = pb + j;
    const float* xb = x + (size_t)b * C_CH * NNODES + (size_t)p * 64;
    float*       ob = out + (size_t)b * C_CH * NNODES + (size_t)p * 64;
    const float* rw = &s_row[j * ROWPAD];

#pragma unroll
    for (int i = 0; i < 16; ++i) {
      const int e = i * 256 + t;          // 0..4095 within tile
      const int c = e >> 6;               // channel (uniform per wave)
      const int q = e & 63;               // node-in-block (lane-contiguous)

      // ---- basis from own element (coalesced global read) ----
      const float xv = xb[(size_t)c * NNODES + q];
      int id = (int)((xv + 1.0f) * 16.0f);            // trunc toward zero
      id = id < 0 ? 0 : (id > SEGMENTS - 1 ? SEGMENTS - 1 : id);
      const float tt = fmaf(32.0f, xv, (float)(31 - 2 * id));  // local coord [-1,1]
      const float p0 = tt + 1.0f;
      const float p1 = tt + (1.0f / 3.0f);
      const float p2 = tt - (1.0f / 3.0f);
      const float p3 = tt - 1.0f;
      const float b0 = p1 * p2 * p3 * (-0.5625f);     //  -9/16
      const float b1 = p0 * p2 * p3 * 1.6875f;        //  27/16
      const float b2 = p0 * p1 * p3 * (-1.6875f);     // -27/16
      const float b3 = p0 * p1 * p2 * 0.5625f;        //   9/16

      // ---- bin index from permuted element x[b, p, q*64 + c] ----
      const float xp = rw[q * 65 + c];
      int ip = (int)((xp + 1.0f) * 16.0f);
      ip = ip < 0 ? 0 : (ip > SEGMENTS - 1 ? SEGMENTS - 1 : ip);

      // w-row gather: c uniform per wave; bank hit needs 3*(idx-idx') == 0 mod 64
      // -> only equal addresses collide, which LDS broadcasts. Conflict-free.
      const float* wr = &s_w[c * WROW + 3 * ip];
      float acc = b0 * wr[0];
      acc = fmaf(b1, wr[1], acc);
      acc = fmaf(b2, wr[2], acc);
      acc = fmaf(b3, wr[3], acc);

      ob[(size_t)c * NNODES + q] = acc;   // coalesced store
    }
  }
}

extern "C" void kernel_launch(void* const* d_in, const int* in_sizes, int n_in,
                              void* d_out, int out_size, void* d_ws, size_t ws_size,
                              hipStream_t stream) {
  const float* x = (const float*)d_in[0];
  const float* w = (const float*)d_in[1];
  float* out = (float*)d_out;
  dim3 grid(NNODES / 64 / NP, BATCH);     // 32 x 32 = 1024 workgroups
  PiecewiseShared_70952859730191_kernel<<<grid, 256, 0, stream>>>(x, w, out);
}